// WeightList_44573170598308
// MI455X (gfx1250) — compile-verified
//
#include <hip/hip_runtime.h>
#include <math.h>

typedef __attribute__((ext_vector_type(2))) float v2f;
typedef __attribute__((ext_vector_type(8))) float v8f;

#define DIMK   128      // embedding dim (reference: 128)
#define KSTEPS (DIMK / 4)
#define TILE   16
#define WAVES  8
#define BLOCK  (WAVES * 32)
#define PITCH  132      // LDS pitch in floats (pad to avoid 16-way bank conflict)
#define BIGF   1e30f

#define ALPHA_ 10.0f
#define BETA_  2.0f
#define BASE_  0.5f

__device__ __forceinline__ float half_min16(float v) {
#pragma unroll
    for (int m = 8; m >= 1; m >>= 1) v = fminf(v, __shfl_xor(v, m, 32));
    return v;
}
__device__ __forceinline__ float half_max16(float v) {
#pragma unroll
    for (int m = 8; m >= 1; m >>= 1) v = fmaxf(v, __shfl_xor(v, m, 32));
    return v;
}
__device__ __forceinline__ float half_sum16(float v) {
#pragma unroll
    for (int m = 8; m >= 1; m >>= 1) v += __shfl_xor(v, m, 32);
    return v;
}
__device__ __forceinline__ int half_isum16(int v) {
#pragma unroll
    for (int m = 8; m >= 1; m >>= 1) v += __shfl_xor(v, m, 32);
    return v;
}

// Preload this lane's A operand (16x4 fp32 fragments for all 32 K-steps) from
// the LDS row tile into registers: 64 VGPRs, loop-invariant across all column
// tiles of the workgroup.
__device__ __forceinline__ void load_a_regs(const float* __restrict__ a_lds,
                                            int l16, int half, v2f (&a)[KSTEPS]) {
    const float* aptr = a_lds + l16 * PITCH + half * 2;
#pragma unroll
    for (int k = 0; k < KSTEPS; ++k) a[k] = *(const v2f*)(aptr + 4 * k);
}

// Two 16x16 sim tiles: shared register-resident A, two independent B streams ->
// two independent WMMA accumulation chains (a single chain is RAW-serialized
// per ISA 7.12.1; interleaving keeps the XDL pipe fed).
// C/D layout: lane l, component v -> sim[rowBase + v + 8*(l>>4)][colBase + (l&15)]
__device__ __forceinline__ void sim_tile2(const v2f (&a)[KSTEPS],
                                          const float* __restrict__ x,
                                          int colBase0, int colBase1,
                                          int l16, int half,
                                          v8f& c0, v8f& c1) {
    c0 = (v8f){}; c1 = (v8f){};
    const float* b0 = x + (size_t)(colBase0 + l16) * DIMK + half * 2; // B: 4x16 (X^T tile)
    const float* b1 = x + (size_t)(colBase1 + l16) * DIMK + half * 2;
#pragma unroll
    for (int k = 0; k < KSTEPS; ++k) {
        v2f bb0 = *(const v2f*)(b0 + 4 * k);
        v2f bb1 = *(const v2f*)(b1 + 4 * k);
        c0 = __builtin_amdgcn_wmma_f32_16x16x4_f32(false, a[k], false, bb0,
                                                   (short)0, c0, false, false);
        c1 = __builtin_amdgcn_wmma_f32_16x16x4_f32(false, a[k], false, bb1,
                                                   (short)0, c1, false, false);
    }
}

__device__ __forceinline__ v8f sim_tile1(const v2f (&a)[KSTEPS],
                                         const float* __restrict__ x,
                                         int colBase, int l16, int half) {
    v8f c = (v8f){};
    const float* bptr = x + (size_t)(colBase + l16) * DIMK + half * 2;
#pragma unroll
    for (int k = 0; k < KSTEPS; ++k) {
        v2f b = *(const v2f*)(bptr + 4 * k);
        c = __builtin_amdgcn_wmma_f32_16x16x4_f32(false, a[k], false, b,
                                                  (short)0, c, false, false);
    }
    return c;
}

__global__ void ms_init(float* out) { out[0] = 0.0f; }

__global__ __launch_bounds__(BLOCK)
void ms_pass1(const float* __restrict__ x, const int* __restrict__ targets,
              float* __restrict__ min_pos, float* __restrict__ max_neg, int n) {
    __shared__ float a_lds[TILE * PITCH];
    __shared__ int   tgt_row[TILE];
    __shared__ float red_min[WAVES][TILE];
    __shared__ float red_max[WAVES][TILE];

    const int tid  = threadIdx.x;
    const int wave = tid >> 5, lane = tid & 31;
    const int half = lane >> 4, l16 = lane & 15;
    const int rowBase = blockIdx.x * TILE;

    for (int idx = tid; idx < TILE * DIMK; idx += BLOCK) {
        int r = idx / DIMK, ccol = idx % DIMK;
        a_lds[r * PITCH + ccol] = x[(size_t)(rowBase + r) * DIMK + ccol];
    }
    if (tid < TILE) tgt_row[tid] = targets[rowBase + tid];
    __syncthreads();

    v2f areg[KSTEPS];
    load_a_regs(a_lds, l16, half, areg);

    int ti[8];
#pragma unroll
    for (int v = 0; v < 8; ++v) ti[v] = tgt_row[v + 8 * half];

    float mn[8], mx[8];
#pragma unroll
    for (int v = 0; v < 8; ++v) { mn[v] = BIGF; mx[v] = -BIGF; }

    const int ntiles   = n / TILE;
    const int per_wave = ntiles / WAVES;   // assumes ntiles % WAVES == 0 (n=4096)
    const int npairs   = per_wave >> 1;    // scalar trip count -> s_cbranch loop

    for (int p = 0; p < npairs; ++p) {
        const int t0 = (2 * p) * WAVES + wave;
        const int t1 = t0 + WAVES;
        if (p + 1 < npairs)   // speculative prefetch of next pair's B rows
            __builtin_prefetch(x + (size_t)((t0 + 2 * WAVES) * TILE + l16) * DIMK, 0, 0);

        v8f c0, c1;
        sim_tile2(areg, x, t0 * TILE, t1 * TILE, l16, half, c0, c1);
        const int tj0 = targets[t0 * TILE + l16];
        const int tj1 = targets[t1 * TILE + l16];
#pragma unroll
        for (int v = 0; v < 8; ++v) {
            float s0 = c0[v], s1 = c1[v];
            if (ti[v] == tj0) { if (s0 < 1.0f) mn[v] = fminf(mn[v], s0); }
            else              { mx[v] = fmaxf(mx[v], s0); }
            if (ti[v] == tj1) { if (s1 < 1.0f) mn[v] = fminf(mn[v], s1); }
            else              { mx[v] = fmaxf(mx[v], s1); }
        }
    }
    if (per_wave & 1) {   // scalar-uniform epilogue (not taken for n=4096)
        const int t = (per_wave - 1) * WAVES + wave;
        v8f c = sim_tile1(areg, x, t * TILE, l16, half);
        const int tj = targets[t * TILE + l16];
#pragma unroll
        for (int v = 0; v < 8; ++v) {
            float s = c[v];
            if (ti[v] == tj) { if (s < 1.0f) mn[v] = fminf(mn[v], s); }
            else             { mx[v] = fmaxf(mx[v], s); }
        }
    }

#pragma unroll
    for (int v = 0; v < 8; ++v) { mn[v] = half_min16(mn[v]); mx[v] = half_max16(mx[v]); }
    if (l16 == 0) {
#pragma unroll
        for (int v = 0; v < 8; ++v) {
            red_min[wave][v + 8 * half] = mn[v];
            red_max[wave][v + 8 * half] = mx[v];
        }
    }
    __syncthreads();
    if (tid < TILE) {
        float a = BIGF, b = -BIGF;
#pragma unroll
        for (int w = 0; w < WAVES; ++w) {
            a = fminf(a, red_min[w][tid]);
            b = fmaxf(b, red_max[w][tid]);
        }
        min_pos[rowBase + tid] = a;
        max_neg[rowBase + tid] = b;
    }
}

__device__ __forceinline__ void accum_tile(const v8f& c, int tj, int hm,
                                           const int (&ti)[8], const float (&mar)[8],
                                           const float (&wgt)[8], const float (&mp)[8],
                                           const float (&mxn)[8],
                                           float (&ps)[8], float (&ns)[8],
                                           int (&apc)[8], int (&anc)[8]) {
#pragma unroll
    for (int v = 0; v < 8; ++v) {
        float s = c[v];
        bool same = (ti[v] == tj);
        if (hm == 1) {
            float ws = s * wgt[v];
            if (same) {
                if (s < 1.0f && (mxn[v] - s + mar[v] > 0.0f)) {
                    ps[v] += __expf(-BETA_ * (ws - BASE_)); apc[v]++;
                }
            } else if (s + mar[v] - mp[v] > 0.0f) {
                ns[v] += __expf(ALPHA_ * (ws - BASE_)); anc[v]++;
            }
        } else {
            if (same) {
                if (s < 1.0f) { ps[v] += __expf(-BETA_ * (s - BASE_ - mar[v])); apc[v]++; }
            } else {
                ns[v] += __expf(ALPHA_ * (s - BASE_ + mar[v])); anc[v]++;
            }
        }
    }
}

__global__ __launch_bounds__(BLOCK)
void ms_pass2(const float* __restrict__ x, const int* __restrict__ targets,
              const float* __restrict__ margin, const float* __restrict__ weight,
              const int* __restrict__ hm_ptr,
              const float* __restrict__ min_pos, const float* __restrict__ max_neg,
              float* __restrict__ out, int n) {
    __shared__ float a_lds[TILE * PITCH];
    __shared__ int   tgt_row[TILE];
    __shared__ float mar_row[TILE], wgt_row[TILE], mp_row[TILE], mx_row[TILE];
    __shared__ float red_ps[WAVES][TILE], red_ns[WAVES][TILE];
    __shared__ int   red_ap[WAVES][TILE], red_an[WAVES][TILE];

    const int tid  = threadIdx.x;
    const int wave = tid >> 5, lane = tid & 31;
    const int half = lane >> 4, l16 = lane & 15;
    const int rowBase = blockIdx.x * TILE;
    const int hm = hm_ptr[0];

    for (int idx = tid; idx < TILE * DIMK; idx += BLOCK) {
        int r = idx / DIMK, ccol = idx % DIMK;
        a_lds[r * PITCH + ccol] = x[(size_t)(rowBase + r) * DIMK + ccol];
    }
    if (tid < TILE) {
        tgt_row[tid] = targets[rowBase + tid];
        mar_row[tid] = margin[rowBase + tid];
        wgt_row[tid] = weight[rowBase + tid];
        mp_row[tid]  = min_pos[rowBase + tid];
        mx_row[tid]  = max_neg[rowBase + tid];
    }
    __syncthreads();

    v2f areg[KSTEPS];
    load_a_regs(a_lds, l16, half, areg);

    int ti[8]; float mar[8], wgt[8], mp[8], mxn[8];
#pragma unroll
    for (int v = 0; v < 8; ++v) {
        int m = v + 8 * half;
        ti[v] = tgt_row[m]; mar[v] = mar_row[m]; wgt[v] = wgt_row[m];
        mp[v] = mp_row[m];  mxn[v] = mx_row[m];
    }

    float ps[8], ns[8]; int apc[8], anc[8];
#pragma unroll
    for (int v = 0; v < 8; ++v) { ps[v] = 0.f; ns[v] = 0.f; apc[v] = 0; anc[v] = 0; }

    const int ntiles   = n / TILE;
    const int per_wave = ntiles / WAVES;
    const int npairs   = per_wave >> 1;

    for (int p = 0; p < npairs; ++p) {
        const int t0 = (2 * p) * WAVES + wave;
        const int t1 = t0 + WAVES;
        if (p + 1 < npairs)
            __builtin_prefetch(x + (size_t)((t0 + 2 * WAVES) * TILE + l16) * DIMK, 0, 0);

        v8f c0, c1;
        sim_tile2(areg, x, t0 * TILE, t1 * TILE, l16, half, c0, c1);
        const int tj0 = targets[t0 * TILE + l16];
        const int tj1 = targets[t1 * TILE + l16];
        accum_tile(c0, tj0, hm, ti, mar, wgt, mp, mxn, ps, ns, apc, anc);
        accum_tile(c1, tj1, hm, ti, mar, wgt, mp, mxn, ps, ns, apc, anc);
    }
    if (per_wave & 1) {
        const int t = (per_wave - 1) * WAVES + wave;
        v8f c = sim_tile1(areg, x, t * TILE, l16, half);
        const int tj = targets[t * TILE + l16];
        accum_tile(c, tj, hm, ti, mar, wgt, mp, mxn, ps, ns, apc, anc);
    }

#pragma unroll
    for (int v = 0; v < 8; ++v) {
        ps[v] = half_sum16(ps[v]);    ns[v] = half_sum16(ns[v]);
        apc[v] = half_isum16(apc[v]); anc[v] = half_isum16(anc[v]);
    }
    if (l16 == 0) {
#pragma unroll
        for (int v = 0; v < 8; ++v) {
            int m = v + 8 * half;
            red_ps[wave][m] = ps[v]; red_ns[wave][m] = ns[v];
            red_ap[wave][m] = apc[v]; red_an[wave][m] = anc[v];
        }
    }
    __syncthreads();
    if (tid < TILE) {
        float psum = 0.f, nsum = 0.f; int ap = 0, an = 0;
#pragma unroll
        for (int w = 0; w < WAVES; ++w) {
            psum += red_ps[w][tid]; nsum += red_ns[w][tid];
            ap += red_ap[w][tid];   an += red_an[w][tid];
        }
        float loss_i = (2.0f / BETA_) * log1pf(psum) + (2.0f / ALPHA_) * log1pf(nsum);
        if (hm == 1) {
            bool valid = (ap + an) >= 1;
            if (!valid) { loss_i = 0.0f; ap = 0; an = 0; }
        }
        const int i = rowBase + tid;
        out[1 + i]     = (float)ap;
        out[1 + n + i] = (float)an;
        atomicAdd(out, loss_i / (float)n);
    }
}

extern "C" void kernel_launch(void* const* d_in, const int* in_sizes, int n_in,
                              void* d_out, int out_size, void* d_ws, size_t ws_size,
                              hipStream_t stream) {
    const float* x       = (const float*)d_in[0];
    const int*   targets = (const int*)d_in[1];
    const float* margin  = (const float*)d_in[2];
    const float* weight  = (const float*)d_in[3];
    const int*   hm      = (const int*)d_in[4];
    const int n = in_sizes[1];                 // 4096
    float* minp = (float*)d_ws;
    float* maxn = minp + n;
    float* out  = (float*)d_out;

    ms_init<<<1, 1, 0, stream>>>(out);
    const int blocks = n / TILE;               // 256 workgroups, 16 rows each
    ms_pass1<<<blocks, BLOCK, 0, stream>>>(x, targets, minp, maxn, n);
    ms_pass2<<<blocks, BLOCK, 0, stream>>>(x, targets, margin, weight, hm,
                                           minp, maxn, out, n);
}